// PiecewiseQuadraticCoupling_21638045237435
// MI455X (gfx1250) — compile-verified
//
#include <hip/hip_runtime.h>

// Piecewise-quadratic coupling forward for MI455X (gfx1250, wave32).
// Memory-bound (~285 MB @ 23.3 TB/s => ~12us floor). The two per-row cumsums
// (edges = cumsum(W), cum_areas = cumsum(0.5*(V[i]+V[i+1])*W[i])) are computed
// as fp32 matmuls against a constant lower-triangular ones matrix using
// V_WMMA_F32_16X16X4_F32, keeping full fp32 precision while offloading the
// scan chains to the matrix pipes. Everything else is coalesced loads, LDS
// staging (stride-33 => bank-conflict-free column access), and a per-lane
// bin search + polynomial finish.

typedef __attribute__((ext_vector_type(2))) float v2f;
typedef __attribute__((ext_vector_type(8))) float v8f;

#define NB 32
#define WAVES_PER_BLOCK 4
#define ROWS_PER_WAVE 32
#define ROWS_PER_BLOCK (WAVES_PER_BLOCK * ROWS_PER_WAVE)
#define VSTRIDE 33   // odd stride: 32 rows map to 32 distinct LDS banks

__global__ __launch_bounds__(128, 1) void pqc_fwd_kernel(
    const float* __restrict__ xb,   // [n]
    const float* __restrict__ Wg,   // [n,32]
    const float* __restrict__ Vg,   // [n,33]
    float* __restrict__ out,        // [2*n]: z then log_det
    int n)
{
  __shared__ float sE [WAVES_PER_BLOCK][ROWS_PER_WAVE * VSTRIDE]; // W, then edges
  __shared__ float sV [WAVES_PER_BLOCK][ROWS_PER_WAVE * VSTRIDE]; // V rows (33 used)
  __shared__ float sCA[WAVES_PER_BLOCK][ROWS_PER_WAVE * VSTRIDE]; // inclusive cum-areas

  const int lane    = threadIdx.x & 31;
  const int w       = threadIdx.x >> 5;
  const int rowBase = (blockIdx.x * WAVES_PER_BLOCK + w) * ROWS_PER_WAVE;
  const int maxRow  = n - 1;

  float* sWf  = sE[w];
  float* sVf  = sV[w];
  float* sCAf = sCA[w];

  // ---- Phase 1: stage W and V into LDS (coalesced) -------------------------
  #pragma unroll
  for (int it = 0; it < 32; ++it) {
    int idx  = it * 32 + lane;                 // 0..1023 over 32x32 tile
    int r    = idx >> 5, c = idx & 31;
    int grow = min(rowBase + r, maxRow);
    sWf[r * VSTRIDE + c] = Wg[(size_t)grow * NB + c];
  }
  const size_t vBase = (size_t)rowBase * 33;   // V global stride == 33 == VSTRIDE
  const size_t vMax  = (size_t)n * 33 - 1;
  #pragma unroll
  for (int it = 0; it < 33; ++it) {
    size_t g = vBase + (size_t)(it * 32 + lane);
    if (g > vMax) g = vMax;
    sVf[it * 32 + lane] = Vg[g];
  }
  __syncthreads();

  // ---- Phase 2: both cumsums via fp32 WMMA (D = A * Tri + C) ---------------
  // A (16x4 f32): lane L in [0,16): M=L, holds K=k0,k0+1; lanes 16..31: K=2,3.
  // B (4x16 f32): lane%16 = N, same K split; B[k][n] = (k_global <= n) ? 1 : 0.
  const int m    = lane & 15;
  const int k0   = (lane >> 4) << 1;           // 0 or 2
  const int ncol = lane & 15;
  const float* wRow = &sWf[m * VSTRIDE];
  const float* vRow = &sVf[m * VSTRIDE];

  v8f cE0 = {}, cE1 = {}, cA0 = {}, cA1 = {};
  #pragma unroll
  for (int c = 0; c < 8; ++c) {
    const int i0 = c * 4 + k0;
    const int i1 = i0 + 1;
    v2f aW, aA, b0, b1;
    aW.x = wRow[i0];
    aW.y = wRow[i1];
    aA.x = 0.5f * (vRow[i0] + vRow[i0 + 1]) * aW.x;   // trapezoid areas
    aA.y = 0.5f * (vRow[i1] + vRow[i1 + 1]) * aW.y;
    b0.x = (i0 <= ncol)      ? 1.0f : 0.0f;           // cols 0..15
    b0.y = (i1 <= ncol)      ? 1.0f : 0.0f;
    b1.x = (i0 <= ncol + 16) ? 1.0f : 0.0f;           // cols 16..31
    b1.y = (i1 <= ncol + 16) ? 1.0f : 0.0f;
    cE0 = __builtin_amdgcn_wmma_f32_16x16x4_f32(false, aW, false, b0, (short)0, cE0, false, false);
    cE1 = __builtin_amdgcn_wmma_f32_16x16x4_f32(false, aW, false, b1, (short)0, cE1, false, false);
    cA0 = __builtin_amdgcn_wmma_f32_16x16x4_f32(false, aA, false, b0, (short)0, cA0, false, false);
    cA1 = __builtin_amdgcn_wmma_f32_16x16x4_f32(false, aA, false, b1, (short)0, cA1, false, false);
  }

  // ---- Phase 3: C fragments -> LDS (16x16 f32 C layout: VGPR r -> M=r/r+8) -
  const int rsel = (lane >> 4) << 3;           // 0 or 8
  #pragma unroll
  for (int r = 0; r < 8; ++r) {
    int mrow = rsel + r;
    sWf [mrow * VSTRIDE + ncol]      = cE0[r];
    sWf [mrow * VSTRIDE + ncol + 16] = cE1[r];
    sCAf[mrow * VSTRIDE + ncol]      = cA0[r];
    sCAf[mrow * VSTRIDE + ncol + 16] = cA1[r];
  }
  __syncthreads();

  // ---- Phase 4: one lane per row: bin search + finish ----------------------
  const int row  = rowBase + lane;
  const int crow = min(row, maxRow);
  float val = xb[crow];
  val = fminf(fmaxf(val, 0.0f), 0.9999f);

  const float* eRow  = &sWf [lane * VSTRIDE];  // eRow[j] = edges[j+1] (inclusive cumsum)
  const float* caRow = &sCAf[lane * VSTRIDE];  // caRow[j] = sum_{i<=j} areas[i]
  const float* vR    = &sVf [lane * VSTRIDE];

  // bin = #{k in 1..31 : val >= edges[k]}  (edges[0]=0 always passes,
  // edges[32] forced to 1.0 never passes since val <= 0.9999)
  int bin = 0;
  #pragma unroll
  for (int j = 0; j < 31; ++j) bin += (val >= eRow[j]) ? 1 : 0;

  float edge_b   = (bin == 0) ? 0.0f : eRow[bin - 1];
  float area_pre = (bin == 0) ? 0.0f : caRow[bin - 1];
  float w_b  = Wg[(size_t)crow * NB + bin];    // exact width (L2/WGP$-hot reread)
  float v_b  = vR[bin];
  float v_b1 = vR[bin + 1];

  float alpha = (val - edge_b) / (w_b + 1e-8f);
  float dv    = v_b1 - v_b;
  float z     = area_pre + alpha * v_b * w_b + 0.5f * alpha * alpha * dv * w_b;
  float ld    = logf(v_b + alpha * dv + 1e-8f);

  if (row < n) {
    out[row]             = z;
    out[(size_t)n + row] = ld;
  }
}

extern "C" void kernel_launch(void* const* d_in, const int* in_sizes, int n_in,
                              void* d_out, int out_size, void* d_ws, size_t ws_size,
                              hipStream_t stream) {
  const float* xb = (const float*)d_in[0];   // [n,1] fp32
  const float* W  = (const float*)d_in[1];   // [n,32] fp32
  const float* V  = (const float*)d_in[2];   // [n,33] fp32
  float* out      = (float*)d_out;           // [2*n] fp32: z, then log_det
  const int n     = in_sizes[0];

  const int blocks = (n + ROWS_PER_BLOCK - 1) / ROWS_PER_BLOCK;
  hipLaunchKernelGGL(pqc_fwd_kernel, dim3(blocks), dim3(ROWS_PER_BLOCK / ROWS_PER_WAVE * 32),
                     0, stream, xb, W, V, out, n);
}